// BatchedFasterRCNN_56307021250955
// MI455X (gfx1250) — compile-verified
//
#include <hip/hip_runtime.h>
#include <hip/hip_bf16.h>

// BatchedFasterRCNN post-process for MI455X (gfx1250, wave32).
// One 256-thread block (8 waves) per batch image:
//   Phase 0: async global->LDS staging of boxes/scores/labels (ASYNCcnt path)
//   Phase 1: single pass over 1000 boxes; per-bin top-3 via ds_max_rtn_u64
//            cascade on 64-bit keys (score_bits<<32 | ~index) -- deterministic,
//            matches jax.lax.top_k ordering (score desc, index asc).
//   Phase 2: 147 threads gather winners from LDS and write both outputs.

#define NBOX  1000
#define RESX  7
#define R2    49
#define KDET  3
#define BLOCK 256

typedef int v4i __attribute__((vector_size(16)));

#if defined(__has_builtin)
#  if __has_builtin(__builtin_amdgcn_global_load_async_to_lds_b128)
#    define HAVE_ASYNC_LDS 1
#  endif
#  if __has_builtin(__builtin_amdgcn_s_wait_asynccnt)
#    define HAVE_WAIT_ASYNC 1
#  endif
#endif

__device__ __forceinline__ void async_copy_b128(const void* gptr, void* lptr) {
#if defined(HAVE_ASYNC_LDS)
  // generic->AS1: numeric identity; generic->AS3: low 32 bits are the LDS offset.
  __builtin_amdgcn_global_load_async_to_lds_b128(
      (__attribute__((address_space(1))) v4i*)(unsigned long long)gptr,
      (__attribute__((address_space(3))) v4i*)(unsigned int)(unsigned long long)lptr,
      0, 0);
#else
  *(float4*)lptr = *(const float4*)gptr;
#endif
}

__device__ __forceinline__ void wait_async_copies() {
#if defined(HAVE_ASYNC_LDS)
#  if defined(HAVE_WAIT_ASYNC)
  __builtin_amdgcn_s_wait_asynccnt(0);
#  else
  asm volatile("s_wait_asynccnt 0" ::: "memory");
#  endif
#endif
}

__global__ __launch_bounds__(BLOCK)
void BatchedFasterRCNN_56307021250955_kernel(
    const float* __restrict__ boxes,   // [B,N,4]
    const int*   __restrict__ labels,  // [B,N]
    const float* __restrict__ scores,  // [B,N]
    const int*   __restrict__ Hp,      // scalar
    const int*   __restrict__ Wp,      // scalar
    float*       __restrict__ out,     // classes [B,3,7,7] ++ boxes [B,15,7,7]
    int B)
{
  __shared__ __align__(16) float sBoxes[NBOX * 4];
  __shared__ __align__(16) float sScores[NBOX];
  __shared__ __align__(16) int   sLabels[NBOX];
  __shared__ unsigned long long  topk[R2 * KDET];

  const int b = blockIdx.x;
  const int t = threadIdx.x;

  const float Hf = (float)Hp[0];
  const float ar = (float)((double)Wp[0] / (double)Hp[0]);  // f32(W/H) like jnp

  const float* gBoxes  = boxes  + (size_t)b * NBOX * 4;
  const float* gScores = scores + (size_t)b * NBOX;
  const int*   gLabels = labels + (size_t)b * NBOX;

  // ---- Phase 0: stage this batch into LDS (async, 16B granules) ----
  for (int i = t; i < NBOX; i += BLOCK)
    async_copy_b128(gBoxes + i * 4, &sBoxes[i * 4]);
  for (int i = t; i < NBOX / 4; i += BLOCK)
    async_copy_b128(gScores + i * 4, &sScores[i * 4]);
  for (int i = t; i < NBOX / 4; i += BLOCK)
    async_copy_b128(gLabels + i * 4, &sLabels[i * 4]);

  if (t < R2 * KDET) topk[t] = 0ULL;  // sentinel: empty (real keys have score>0.12)

  wait_async_copies();
  __syncthreads();

  // Bin edges, bit-identical to jnp: bins[j] = (j+1)/7 in f32, x-edges = ar*bins.
  float ex[RESX - 1], ey[RESX - 1];
#pragma unroll
  for (int j = 0; j < RESX - 1; ++j) {
    float bj = (float)(j + 1) / (float)RESX;
    ey[j] = bj;
    ex[j] = ar * bj;
  }

  // ---- Phase 1: bin + concurrent top-3 insert ----
  for (int i = t; i < NBOX; i += BLOCK) {
    float s = sScores[i];
    if (!(s > 0.12f)) continue;
    float nx1 = sBoxes[i * 4 + 0] / Hf;
    float ny1 = sBoxes[i * 4 + 1] / Hf;
    float nx2 = sBoxes[i * 4 + 2] / Hf;
    float ny2 = sBoxes[i * 4 + 3] / Hf;
    float cx = (nx1 + nx2) * 0.5f;
    float cy = (ny1 + ny2) * 0.5f;
    int px = 0, py = 0;
#pragma unroll
    for (int j = 0; j < RESX - 1; ++j) {
      px += (cx >= ex[j]);
      py += (cy >= ey[j]);
    }
    int bin = py * RESX + px;
    // key: score-major desc, index-minor asc (unique per box -> deterministic)
    unsigned long long v = ((unsigned long long)__float_as_uint(s) << 32)
                         | (unsigned long long)(~(unsigned int)i);
#pragma unroll
    for (int slot = 0; slot < KDET; ++slot) {
      unsigned long long old = atomicMax(&topk[bin * KDET + slot], v);
      if (old == 0ULL) break;          // installed into empty slot, nothing displaced
      v = (old < v) ? old : v;         // carry displaced (or rejected) value down
    }
  }

  __syncthreads();

  // ---- Phase 2: emit 147 (bin,k) results per batch ----
  if (t < R2 * KDET) {
    int bin = t / KDET;
    int k   = t - bin * KDET;
    unsigned long long key = topk[bin * KDET + k];

    float cls, f0, f1, f2, f3, f4;
    if (key == 0ULL) {
      cls = 0.0f;
      f0 = f1 = f2 = f3 = f4 = -1.0f;   // res_boxes init
    } else {
      int   i = (int)(~(unsigned int)(key & 0xFFFFFFFFull));
      float s = __uint_as_float((unsigned int)(key >> 32));
      float nx1 = sBoxes[i * 4 + 0] / Hf;
      float ny1 = sBoxes[i * 4 + 1] / Hf;
      float nx2 = sBoxes[i * 4 + 2] / Hf;
      float ny2 = sBoxes[i * 4 + 3] / Hf;
      f0 = s;
      f1 = (nx2 - nx1) / ar;
      f2 = ny2 - ny1;
      f3 = nx1 / ar;
      f4 = ny1;
      cls = (float)sLabels[i];
    }

    // classes_out: [B,3,7,7] -> b*147 + k*49 + bin
    out[(size_t)b * (KDET * R2) + (size_t)k * R2 + bin] = cls;

    // boxes_out: [B,15,7,7], channel = k*5+f
    float* outB = out + (size_t)B * (KDET * R2);
    size_t base = (size_t)b * (KDET * 5 * R2) + (size_t)(k * 5) * R2 + (size_t)bin;
    outB[base + 0 * R2] = f0;
    outB[base + 1 * R2] = f1;
    outB[base + 2 * R2] = f2;
    outB[base + 3 * R2] = f3;
    outB[base + 4 * R2] = f4;
  }
}

extern "C" void kernel_launch(void* const* d_in, const int* in_sizes, int n_in,
                              void* d_out, int out_size, void* d_ws, size_t ws_size,
                              hipStream_t stream) {
  (void)n_in; (void)out_size; (void)d_ws; (void)ws_size;
  const float* boxes  = (const float*)d_in[0];
  const int*   labels = (const int*)d_in[1];
  const float* scores = (const float*)d_in[2];
  const int*   Hp     = (const int*)d_in[3];
  const int*   Wp     = (const int*)d_in[4];
  float* out = (float*)d_out;

  const int B = in_sizes[2] / NBOX;  // pred_scores is [B, N=1000]

  BatchedFasterRCNN_56307021250955_kernel<<<dim3(B), dim3(BLOCK), 0, stream>>>(
      boxes, labels, scores, Hp, Wp, out, B);
}